// RelationalAttentionLayer_19791209300468
// MI455X (gfx1250) — compile-verified
//
#include <hip/hip_runtime.h>
#include <hip/hip_bf16.h>

// ---------------- problem constants (from reference) ----------------
#define NN  10000      // nodes
#define EE  160000     // edges
#define DD  128        // model dim
#define HH  2          // heads
#define RR  8          // relation types
#define HD  256        // H*D
#define EPS 1e-10f

typedef __attribute__((ext_vector_type(2))) float v2f;
typedef __attribute__((ext_vector_type(8))) float v8f;

// ---------------------------------------------------------------
// Generic NT fp32 WMMA GEMM:  C[m,n] = sum_k A[m,k] * B[n,k]
// A row-major [M,K] (lda=K), B row-major [Ncols,K] (ldb=K),
// C row-major [M,Ncols] (ldc=Ncols). blockIdx.z batches B/C.
// One wave computes a 16x64 C super-tile (4 WMMA column tiles sharing
// the A operand). K is a compile-time constant; the K-loop is fully
// unrolled and explicitly software-pipelined (operands for step k+4
// are in flight while the WMMAs of step k issue) so the scheduler can
// overlap global_load_b64 latency with v_wmma issue instead of
// draining loadcnt to 0 before each WMMA.
// Requires M % 16 == 0 and Ncols % 64 == 0 (true for all uses here).
//
// V_WMMA_F32_16X16X4_F32 operand layout (ISA 7.12.2):
//  A 16x4: lane l(0..15) holds {K=0,K=1} for M=l; lane l+16 holds {K=2,K=3}
//  B mirrors with N on lanes. C/D: VGPR i -> M=i (lanes 0-15) / M=i+8
//  (lanes 16-31), N = lane&15.
// ---------------------------------------------------------------
template<int K>
__global__ __launch_bounds__(256)
void gemm_nt_f32_wmma(const float* __restrict__ A, int lda,
                      const float* __restrict__ B, int ldb, long strideB,
                      float* __restrict__ C, int ldc, long strideC,
                      int M, int Ncols)
{
    const float* Bz = B + (long)blockIdx.z * strideB;
    float*       Cz = C + (long)blockIdx.z * strideC;

    const int tilesN = Ncols >> 6;                  // 64-wide super-tiles
    const int nTiles = (M >> 4) * tilesN;
    const int wavesPerBlock = blockDim.x >> 5;
    const int wave = blockIdx.x * wavesPerBlock + (threadIdx.x >> 5);
    if (wave >= nTiles) return;                     // whole-wave uniform exit
    const int lane = threadIdx.x & 31;
    const int l15  = lane & 15;

    const int tm = wave / tilesN;
    const int tn = wave % tilesN;                   // super-tile column index

    const int kOff = (lane < 16) ? 0 : 2;           // K sub-pair per half-wave
    const float* arow = A  + (size_t)(tm * 16 + l15) * lda + kOff;
    const float* br0  = Bz + (size_t)(tn * 64 +  0 + l15) * ldb + kOff;
    const float* br1  = Bz + (size_t)(tn * 64 + 16 + l15) * ldb + kOff;
    const float* br2  = Bz + (size_t)(tn * 64 + 32 + l15) * ldb + kOff;
    const float* br3  = Bz + (size_t)(tn * 64 + 48 + l15) * ldb + kOff;

    v8f acc0 = {}, acc1 = {}, acc2 = {}, acc3 = {};

    // prologue: first operand set
    v2f a  = *(const v2f*)(arow);
    v2f b0 = *(const v2f*)(br0);
    v2f b1 = *(const v2f*)(br1);
    v2f b2 = *(const v2f*)(br2);
    v2f b3 = *(const v2f*)(br3);

#pragma unroll
    for (int k = 0; k < K - 4; k += 4) {
        // prefetch next K-step while current WMMAs issue
        v2f an  = *(const v2f*)(arow + k + 4);
        v2f b0n = *(const v2f*)(br0 + k + 4);
        v2f b1n = *(const v2f*)(br1 + k + 4);
        v2f b2n = *(const v2f*)(br2 + k + 4);
        v2f b3n = *(const v2f*)(br3 + k + 4);

        acc0 = __builtin_amdgcn_wmma_f32_16x16x4_f32(
                   false, a, false, b0, (short)0, acc0, false, false);
        acc1 = __builtin_amdgcn_wmma_f32_16x16x4_f32(
                   false, a, false, b1, (short)0, acc1, false, false);
        acc2 = __builtin_amdgcn_wmma_f32_16x16x4_f32(
                   false, a, false, b2, (short)0, acc2, false, false);
        acc3 = __builtin_amdgcn_wmma_f32_16x16x4_f32(
                   false, a, false, b3, (short)0, acc3, false, false);

        a = an; b0 = b0n; b1 = b1n; b2 = b2n; b3 = b3n;
    }
    // epilogue: last K-step
    acc0 = __builtin_amdgcn_wmma_f32_16x16x4_f32(
               false, a, false, b0, (short)0, acc0, false, false);
    acc1 = __builtin_amdgcn_wmma_f32_16x16x4_f32(
               false, a, false, b1, (short)0, acc1, false, false);
    acc2 = __builtin_amdgcn_wmma_f32_16x16x4_f32(
               false, a, false, b2, (short)0, acc2, false, false);
    acc3 = __builtin_amdgcn_wmma_f32_16x16x4_f32(
               false, a, false, b3, (short)0, acc3, false, false);

    // store the 4 C tiles per ISA C/D layout
    const int mBase = tm * 16 + ((lane < 16) ? 0 : 8);
    float* cptr = Cz + (size_t)mBase * ldc + tn * 64 + l15;
#pragma unroll
    for (int i = 0; i < 8; ++i) {
        cptr[(size_t)i * ldc +  0] = acc0[i];
        cptr[(size_t)i * ldc + 16] = acc1[i];
        cptr[(size_t)i * ldc + 32] = acc2[i];
        cptr[(size_t)i * ldc + 48] = acc3[i];
    }
}

// ---------------------------------------------------------------
// zero a float region (grid-stride)
// ---------------------------------------------------------------
__global__ void zero_f32(float* __restrict__ p, long n)
{
    long i = (long)blockIdx.x * blockDim.x + threadIdx.x;
    long stride = (long)gridDim.x * blockDim.x;
    for (; i < n; i += stride) p[i] = 0.0f;
}

// ---------------------------------------------------------------
// Pass 1: one wave per edge. score[e,h] = Q[dst]·Kf[r,src] per head,
// numer = relu(score/16)^2 + eps ; denom[dst,h] += numer (HW f32 atomic)
// lane covers 8 contiguous HD elements: lanes 0-15 -> head 0, 16-31 -> head 1
// ---------------------------------------------------------------
__global__ __launch_bounds__(256)
void edge_scores(const int* __restrict__ src, const int* __restrict__ dst,
                 const int* __restrict__ et,
                 const float* __restrict__ Q, const float* __restrict__ Kf,
                 float* __restrict__ numer, float* __restrict__ denom)
{
    const int e = blockIdx.x * (blockDim.x >> 5) + (threadIdx.x >> 5);
    if (e >= EE) return;
    const int lane = threadIdx.x & 31;

    const int s = src[e], d = dst[e], r = et[e];

    const float* qp = Q  + (size_t)d * HD + lane * 8;
    const float* kp = Kf + ((size_t)r * NN + s) * HD + lane * 8;

    float4 q0 = *(const float4*)(qp);
    float4 q1 = *(const float4*)(qp + 4);
    float4 k0 = *(const float4*)(kp);
    float4 k1 = *(const float4*)(kp + 4);

    float acc = q0.x*k0.x + q0.y*k0.y + q0.z*k0.z + q0.w*k0.w
              + q1.x*k1.x + q1.y*k1.y + q1.z*k1.z + q1.w*k1.w;

    // reduce within each 16-lane half (one head per half)
#pragma unroll
    for (int off = 8; off >= 1; off >>= 1)
        acc += __shfl_xor(acc, off, 32);

    const float score = acc * 0.0625f;          // 1/sqrt(HD=256)
    const float rl = fmaxf(score, 0.0f);
    const float nm = rl * rl + EPS;

    if ((lane & 15) == 0) {
        const int h = lane >> 4;
        numer[(size_t)e * HH + h] = nm;
        unsafeAtomicAdd(&denom[(size_t)d * HH + h], nm);
    }
}

// ---------------------------------------------------------------
// Pass 2: one wave per edge. w = numer/denom[dst];
// Z[dst, :] += w[h] * Vf[r, src, :]   (HW f32 atomics)
// ---------------------------------------------------------------
__global__ __launch_bounds__(256)
void edge_scatter(const int* __restrict__ src, const int* __restrict__ dst,
                  const int* __restrict__ et,
                  const float* __restrict__ Vf,
                  const float* __restrict__ numer, const float* __restrict__ denom,
                  float* __restrict__ Z)
{
    const int e = blockIdx.x * (blockDim.x >> 5) + (threadIdx.x >> 5);
    if (e >= EE) return;
    const int lane = threadIdx.x & 31;
    const int h = lane >> 4;

    const int s = src[e], d = dst[e], r = et[e];
    const float w = numer[(size_t)e * HH + h] / denom[(size_t)d * HH + h];

    const float* vp = Vf + ((size_t)r * NN + s) * HD + lane * 8;
    float*       zp = Z  + (size_t)d * HD + lane * 8;

    float4 v0 = *(const float4*)(vp);
    float4 v1 = *(const float4*)(vp + 4);

    unsafeAtomicAdd(zp + 0, w * v0.x);
    unsafeAtomicAdd(zp + 1, w * v0.y);
    unsafeAtomicAdd(zp + 2, w * v0.z);
    unsafeAtomicAdd(zp + 3, w * v0.w);
    unsafeAtomicAdd(zp + 4, w * v1.x);
    unsafeAtomicAdd(zp + 5, w * v1.y);
    unsafeAtomicAdd(zp + 6, w * v1.z);
    unsafeAtomicAdd(zp + 7, w * v1.w);
}

// ---------------------------------------------------------------
extern "C" void kernel_launch(void* const* d_in, const int* in_sizes, int n_in,
                              void* d_out, int out_size, void* d_ws, size_t ws_size,
                              hipStream_t stream)
{
    (void)in_sizes; (void)n_in; (void)out_size; (void)ws_size;

    const float* X  = (const float*)d_in[0];   // [N, D]
    const int*   src = (const int*)d_in[1];    // [E]
    const int*   dst = (const int*)d_in[2];    // [E]
    const int*   et  = (const int*)d_in[3];    // [E]
    const float* WQ = (const float*)d_in[4];   // [HD, D]
    const float* WK = (const float*)d_in[5];   // [R, HD, D]
    const float* WV = (const float*)d_in[6];   // [R, HD, D]
    const float* WO = (const float*)d_in[7];   // [D, HD]
    float* out = (float*)d_out;                // [N, D]

    // ---- workspace layout (floats) ----
    float* ws = (float*)d_ws;
    const size_t szQ  = (size_t)NN * HD;            // 2,560,000
    const size_t szKf = (size_t)RR * NN * HD;       // 20,480,000
    const size_t szNm = (size_t)EE * HH;            // 320,000
    const size_t szDn = (size_t)NN * HH;            // 20,000
    float* Q     = ws;
    float* Kf    = Q  + szQ;
    float* Vf    = Kf + szKf;
    float* numer = Vf + szKf;
    float* denom = numer + szNm;
    float* Z     = denom + szDn;                    // [N, HD], follows denom

    // 1) zero accumulators (denom and Z are contiguous)
    {
        long n = (long)(szDn + szQ);                // denom + Z
        int blocks = (int)((n + 255) / 256);
        if (blocks > 4096) blocks = 4096;
        zero_f32<<<blocks, 256, 0, stream>>>(denom, n);
    }

    // 2) Q = X @ WQ^T   (M=N, Ncols=HD, K=D)
    {
        int tiles = (NN / 16) * (HD / 64);          // 2500
        dim3 grid((tiles + 7) / 8, 1, 1);
        gemm_nt_f32_wmma<DD><<<grid, 256, 0, stream>>>(
            X, DD, WQ, DD, 0L, Q, HD, 0L, NN, HD);
    }

    // 3) Kf[r] = X @ WK[r]^T  and  Vf[r] = X @ WV[r]^T   (batched over z=R)
    {
        int tiles = (NN / 16) * (HD / 64);
        dim3 grid((tiles + 7) / 8, 1, RR);
        gemm_nt_f32_wmma<DD><<<grid, 256, 0, stream>>>(
            X, DD, WK, DD, (long)HD * DD, Kf, HD, (long)NN * HD, NN, HD);
        gemm_nt_f32_wmma<DD><<<grid, 256, 0, stream>>>(
            X, DD, WV, DD, (long)HD * DD, Vf, HD, (long)NN * HD, NN, HD);
    }

    // 4) per-edge scores + denom segment-sum
    {
        dim3 grid(EE / 8, 1, 1);                    // 8 waves / block, 1 edge / wave
        edge_scores<<<grid, 256, 0, stream>>>(src, dst, et, Q, Kf, numer, denom);
    }

    // 5) per-edge weighted value scatter into Z
    {
        dim3 grid(EE / 8, 1, 1);
        edge_scatter<<<grid, 256, 0, stream>>>(src, dst, et, Vf, numer, denom, Z);
    }

    // 6) out = Z @ WO^T   (M=N, Ncols=D, K=HD)
    {
        int tiles = (NN / 16) * (DD / 64);          // 1250
        dim3 grid((tiles + 7) / 8, 1, 1);
        gemm_nt_f32_wmma<HD><<<grid, 256, 0, stream>>>(
            Z, HD, WO, HD, 0L, out, DD, 0L, NN, DD);
    }
}